// NodeModel_13125420057115
// MI455X (gfx1250) — compile-verified
//
#include <hip/hip_runtime.h>

// ---------------- problem constants ----------------
#define N_NODES 50000
#define N_EDGES 1000000
#define FX 128            // node feature dim
#define FE 64             // edge feature dim
#define FU 64             // global feature dim
#define KIN 256           // concat width = FX+FE+FU
#define HID 256           // hidden width
#define STRIDE 260        // LDS row stride (floats): 260%64==4 -> conflict-free frag loads

typedef float v2f __attribute__((ext_vector_type(2)));
typedef float v8f __attribute__((ext_vector_type(8)));

// ---------------- kernel 1: zero the scatter workspace ----------------
__global__ void zero_ws_kernel(float4* __restrict__ ws, int n4) {
    int i = blockIdx.x * blockDim.x + threadIdx.x;
    if (i < n4) ws[i] = make_float4(0.f, 0.f, 0.f, 0.f);
}

// ---------------- kernel 2: segment-sum of edge_attr by dest ----------------
// 16 threads per edge, each thread handles 4 contiguous features (float4 load).
__global__ void scatter_edges_kernel(const float* __restrict__ edge_attr,
                                     const int*   __restrict__ edge_index,
                                     float* __restrict__ sums,
                                     float* __restrict__ cnt) {
    int gid  = blockIdx.x * blockDim.x + threadIdx.x;   // < 16M, fits int
    int e    = gid >> 4;
    int part = gid & 15;
    if (e >= N_EDGES) return;
    int d = edge_index[N_EDGES + e];                    // edge_index[1][e] = dest
    float4 v = ((const float4*)edge_attr)[(size_t)e * 16 + part];
    float* s = sums + (size_t)d * FE + part * 4;
    unsafeAtomicAdd(s + 0, v.x);
    unsafeAtomicAdd(s + 1, v.y);
    unsafeAtomicAdd(s + 2, v.z);
    unsafeAtomicAdd(s + 3, v.w);
    if (part == 0) unsafeAtomicAdd(cnt + d, 1.0f);
}

// ---------------- kernel 3: fused mean + concat + 2-layer MLP (f32 WMMA) ----
// Block: 128 threads (4 waves), 32 rows (2 strips of 16). A-tile and H-tile in LDS.
__global__ void __launch_bounds__(128, 1)
fused_concat_mlp_kernel(const float* __restrict__ x,
                        const int*   __restrict__ batch,
                        const float* __restrict__ u,
                        const float* __restrict__ sums,
                        const float* __restrict__ cnt,
                        const float* __restrict__ W1,
                        const float* __restrict__ b1,
                        const float* __restrict__ W2,
                        const float* __restrict__ b2,
                        float* __restrict__ out) {
    extern __shared__ float smem[];
    float* As = smem;                 // [32][STRIDE]
    float* Hs = smem + 32 * STRIDE;   // [32][STRIDE]

    const int tid  = threadIdx.x;
    const int lane = tid & 31;
    const int wave = tid >> 5;
    const int half = lane >> 4;       // 0: lanes 0-15, 1: lanes 16-31
    const int nloc = lane & 15;
    const int row0 = blockIdx.x * 32;

    // ---- Phase 0: A = [x | sums/max(cnt,1) | u[batch]] into LDS; emit edge_out_bar ----
    for (int i = tid; i < 32 * 64; i += 128) {          // 64 float4 per row
        const int r  = i >> 6;
        const int c4 = i & 63;
        const int g  = row0 + r;
        if (g < N_NODES) {
            float4 v;
            if (c4 < 32) {                               // x: cols 0..127
                v = ((const float4*)x)[(size_t)g * 32 + c4];
            } else if (c4 < 48) {                        // mean edges: cols 128..191
                const float inv = 1.0f / fmaxf(cnt[g], 1.0f);
                float4 s = ((const float4*)sums)[(size_t)g * 16 + (c4 - 32)];
                v.x = s.x * inv; v.y = s.y * inv; v.z = s.z * inv; v.w = s.w * inv;
                ((float4*)(out + (size_t)N_NODES * FX))[(size_t)g * 16 + (c4 - 32)] = v;
            } else {                                     // u[batch]: cols 192..255
                const int bg = batch[g];
                v = ((const float4*)u)[bg * 16 + (c4 - 48)];
            }
            float* a = &As[r * STRIDE + c4 * 4];
            a[0] = v.x; a[1] = v.y; a[2] = v.z; a[3] = v.w;
        }
    }
    __syncthreads();

    // ---- Phase 1: H = relu(A @ W1 + b1), V_WMMA_F32_16X16X4_F32 ----
    // A frag: lane m=nloc, K pair at k0+2*half. B frag: K rows k0+half and k0+half+2, N=n.
    for (int nb = wave; nb < HID / 16; nb += 4) {
        const int n = nb * 16 + nloc;
        const float bias = b1[n];
        v8f acc0, acc1;
        #pragma unroll
        for (int j = 0; j < 8; ++j) { acc0[j] = bias; acc1[j] = bias; }
        #pragma unroll 4
        for (int kk = 0; kk < KIN / 4; ++kk) {
            const int k0 = kk * 4;
            v2f b;
            b.x = W1[(size_t)(k0 + half)     * HID + n];
            b.y = W1[(size_t)(k0 + half + 2) * HID + n];
            const float* ap0 = &As[nloc * STRIDE        + k0 + 2 * half];
            const float* ap1 = &As[(16 + nloc) * STRIDE + k0 + 2 * half];
            v2f a0; a0.x = ap0[0]; a0.y = ap0[1];
            v2f a1; a1.x = ap1[0]; a1.y = ap1[1];
            acc0 = __builtin_amdgcn_wmma_f32_16x16x4_f32(false, a0, false, b,
                                                         (short)0, acc0, false, false);
            acc1 = __builtin_amdgcn_wmma_f32_16x16x4_f32(false, a1, false, b,
                                                         (short)0, acc1, false, false);
        }
        #pragma unroll
        for (int j = 0; j < 8; ++j) {                    // C layout: M=j+8*half, N=nloc
            const int m = j + half * 8;
            Hs[m * STRIDE + n]        = fmaxf(acc0[j], 0.0f);
            Hs[(16 + m) * STRIDE + n] = fmaxf(acc1[j], 0.0f);
        }
    }
    __syncthreads();

    // ---- Phase 2: Y = relu(H @ W2 + b2) ----
    for (int nb = wave; nb < FX / 16; nb += 4) {
        const int n = nb * 16 + nloc;
        const float bias = b2[n];
        v8f acc0, acc1;
        #pragma unroll
        for (int j = 0; j < 8; ++j) { acc0[j] = bias; acc1[j] = bias; }
        #pragma unroll 4
        for (int kk = 0; kk < HID / 4; ++kk) {
            const int k0 = kk * 4;
            v2f b;
            b.x = W2[(size_t)(k0 + half)     * FX + n];
            b.y = W2[(size_t)(k0 + half + 2) * FX + n];
            const float* ap0 = &Hs[nloc * STRIDE        + k0 + 2 * half];
            const float* ap1 = &Hs[(16 + nloc) * STRIDE + k0 + 2 * half];
            v2f a0; a0.x = ap0[0]; a0.y = ap0[1];
            v2f a1; a1.x = ap1[0]; a1.y = ap1[1];
            acc0 = __builtin_amdgcn_wmma_f32_16x16x4_f32(false, a0, false, b,
                                                         (short)0, acc0, false, false);
            acc1 = __builtin_amdgcn_wmma_f32_16x16x4_f32(false, a1, false, b,
                                                         (short)0, acc1, false, false);
        }
        #pragma unroll
        for (int j = 0; j < 8; ++j) {
            const int m  = j + half * 8;
            const int g0 = row0 + m;
            const int g1 = row0 + 16 + m;
            if (g0 < N_NODES) out[(size_t)g0 * FX + n] = fmaxf(acc0[j], 0.0f);
            if (g1 < N_NODES) out[(size_t)g1 * FX + n] = fmaxf(acc1[j], 0.0f);
        }
    }
}

// ---------------- launcher ----------------
extern "C" void kernel_launch(void* const* d_in, const int* in_sizes, int n_in,
                              void* d_out, int out_size, void* d_ws, size_t ws_size,
                              hipStream_t stream) {
    const float* x          = (const float*)d_in[0];
    const int*   edge_index = (const int*)  d_in[1];   // [2, E] flattened
    const float* edge_attr  = (const float*)d_in[2];
    const float* u          = (const float*)d_in[3];
    const int*   batch      = (const int*)  d_in[4];
    const float* W1         = (const float*)d_in[5];
    const float* b1         = (const float*)d_in[6];
    const float* W2         = (const float*)d_in[7];
    const float* b2         = (const float*)d_in[8];
    float* out  = (float*)d_out;
    float* sums = (float*)d_ws;                          // [N_NODES*FE] f32 (12.8 MB)
    float* cnt  = sums + (size_t)N_NODES * FE;           // [N_NODES]    f32 (0.2 MB)

    // 1) zero scatter workspace (13 MB) every call — graph-replay safe
    const int n4 = (N_NODES * FE + N_NODES) / 4;         // 812500 float4
    zero_ws_kernel<<<(n4 + 255) / 256, 256, 0, stream>>>((float4*)d_ws, n4);

    // 2) segment-sum + counts via f32 global atomics (sums array is L2-resident)
    const long long sthreads = (long long)N_EDGES * 16;
    scatter_edges_kernel<<<(int)((sthreads + 255) / 256), 256, 0, stream>>>(
        edge_attr, edge_index, sums, cnt);

    // 3) fused mean + concat + MLP with f32 WMMA
    const size_t shmem = (size_t)2 * 32 * STRIDE * sizeof(float); // 66560 B
    hipFuncSetAttribute((const void*)fused_concat_mlp_kernel,
                        hipFuncAttributeMaxDynamicSharedMemorySize, (int)shmem);
    fused_concat_mlp_kernel<<<(N_NODES + 31) / 32, 128, shmem, stream>>>(
        x, batch, u, sums, cnt, W1, b1, W2, b2, out);
}